// CNN_46445776339040
// MI455X (gfx1250) — compile-verified
//
#include <hip/hip_runtime.h>

typedef __attribute__((ext_vector_type(2))) float v2f;
typedef __attribute__((ext_vector_type(8))) float v8f;

#define NWORDS (64 * 256)   // B*S independent words
#define H 20                // word_length
#define E 128               // embedding dim
#define PAD 2

// One wave32 per word. Conv-as-band-matmul on V_WMMA_F32_16X16X4_F32:
//   Y(20x128) = M(20x20 band of conv weights) * X(20x128), then max over rows.
// Two overlapping 16-row M tiles (rowbase 0 and 4) cover all 20 rows; overlap
// is fine because the reduction is max. K=20 -> 5 K-steps of 4.
__global__ __launch_bounds__(256) void conv_max_wmma(
    const float* __restrict__ x,     // (NWORDS, H, E)
    const float* __restrict__ w5,    // 5 conv taps
    const float* __restrict__ bias,  // 1
    float* __restrict__ out)         // (NWORDS, E)
{
    const int lane  = threadIdx.x & 31;
    const int wave  = threadIdx.x >> 5;
    const int word  = blockIdx.x * 8 + wave;

    const int col   = lane & 15;        // A: row M within tile; B: column N (e)
    const int khalf = (lane >> 4) * 2;  // K offset selected by lane half

    float w[5];
#pragma unroll
    for (int i = 0; i < 5; ++i) w[i] = w5[i];
    const float b0 = bias[0];

    // A tiles: band matrix M[h][j] = w[j-h+PAD] if 0<=j-h+PAD<5 else 0.
    // ISA f32 16x4 A layout: lane l holds row M=l%16; VGPR r holds K = (l/16)*2 + r.
    v2f a_rb0[5], a_rb4[5];
#pragma unroll
    for (int ks = 0; ks < 5; ++ks) {
#pragma unroll
        for (int r = 0; r < 2; ++r) {
            const int j  = ks * 4 + khalf + r;
            const int d0 = j - (col + 0) + PAD;
            const int d4 = j - (col + 4) + PAD;
            a_rb0[ks][r] = (d0 >= 0 && d0 < 5) ? w[d0] : 0.0f;
            a_rb4[ks][r] = (d4 >= 0 && d4 < 5) ? w[d4] : 0.0f;
        }
    }

    const float* xw = x + (size_t)word * (H * E);

#pragma unroll 2
    for (int ec = 0; ec < 8; ++ec) {           // 8 chunks of 16 columns = E
        const float* base = xw + ec * 16 + col;
        v8f acc0 = {};                          // rows 0..15
        v8f acc4 = {};                          // rows 4..19
#pragma unroll
        for (int ks = 0; ks < 5; ++ks) {
            const int j0 = ks * 4 + khalf;      // K row pair this lane-half owns
            v2f bb;
            bb[0] = base[(j0 + 0) * E];         // coalesced: 2x64B per load
            bb[1] = base[(j0 + 1) * E];
            acc0 = __builtin_amdgcn_wmma_f32_16x16x4_f32(
                false, a_rb0[ks], false, bb, (short)0, acc0, false, false);
            acc4 = __builtin_amdgcn_wmma_f32_16x16x4_f32(
                false, a_rb4[ks], false, bb, (short)0, acc4, false, false);
        }
        // D layout: lane l, VGPR v -> row (v + 8*(l/16)). Max over all rows
        // this lane covers, then merge lane halves (rows 0..11 U 8..19 = 0..19).
        float m = acc0[0];
#pragma unroll
        for (int v = 1; v < 8; ++v) m = fmaxf(m, acc0[v]);
#pragma unroll
        for (int v = 0; v < 8; ++v) m = fmaxf(m, acc4[v]);
        m = fmaxf(m, __shfl_xor(m, 16, 32));    // wave32 cross-half merge
        if (lane < 16)
            out[(size_t)word * E + ec * 16 + lane] = m + b0;
    }
}

extern "C" void kernel_launch(void* const* d_in, const int* in_sizes, int n_in,
                              void* d_out, int out_size, void* d_ws, size_t ws_size,
                              hipStream_t stream) {
    const float* x    = (const float*)d_in[0];  // (64,256,20,128) f32
    const float* w    = (const float*)d_in[1];  // (1,1,5,1) f32
    const float* b    = (const float*)d_in[2];  // (1,) f32
    float*       out  = (float*)d_out;          // (64,256,128) f32

    dim3 block(256);                // 8 waves -> 8 words per block
    dim3 grid(NWORDS / 8);          // 16384 words
    conv_max_wmma<<<grid, block, 0, stream>>>(x, w, b, out);
}